// GlimpseNetwork_16535624090312
// MI455X (gfx1250) — compile-verified
//
#include <hip/hip_runtime.h>

typedef float v2f __attribute__((ext_vector_type(2)));
typedef float v8f __attribute__((ext_vector_type(8)));

__device__ __forceinline__ v8f wmma_f32(v2f a, v2f b, v8f c) {
    // V_WMMA_F32_16X16X4_F32 : D = A(16x4) * B(4x16) + C(16x16), fp32 throughout
    return __builtin_amdgcn_wmma_f32_16x16x4_f32(
        /*neg_a=*/false, a, /*neg_b=*/false, b,
        /*c_mod=*/(short)0, c, /*reuse_a=*/false, /*reuse_b=*/false);
}

__device__ __forceinline__ int iclamp(int v, int lo, int hi) {
    return v < lo ? lo : (v > hi ? hi : v);
}

// ---------------------------------------------------------------------------
// Foveation: phi[b,p,16,16] = avg-pool of (16<<i)-sized patch at glimpse loc.
// One 256-thread block per (b, p). Also writes bbox = [sy, sx, sy+g, sx+g].
// ---------------------------------------------------------------------------
__global__ void k_foveate(const float* __restrict__ x, const float* __restrict__ l,
                          float* __restrict__ phi, float* __restrict__ bbox) {
    const int bp = blockIdx.x;              // 0 .. 512*12-1
    const int b = bp / 12, p = bp % 12;
    const int j = p / 3, i = p % 3;
    const int gh = 16 << i;                 // 16, 32, 64
    const int k  = 1 << i;                  // pool factor 1, 2, 4

    const float l0 = l[b * 8 + j * 2 + 0];
    const float l1 = l[b * 8 + j * 2 + 1];
    // trunc-toward-zero like .astype(int32), then clamp start inside image
    int sx = (int)(0.5f * ((l0 + 1.0f) * 255.0f) - (float)(gh >> 1));
    int sy = (int)(0.5f * ((l1 + 1.0f) * 255.0f) - (float)(gh >> 1));
    sx = iclamp(sx, 0, 256 - gh);
    sy = iclamp(sy, 0, 256 - gh);

    const int t = threadIdx.x;              // 0..255 -> (gy, gx)
    const int gy = t >> 4, gx = t & 15;
    const float* xb = x + (size_t)b * 65536;

    float s = 0.0f;
    for (int dy = 0; dy < k; ++dy) {
        const float* xr = xb + (sx + gy * k + dy) * 256 + sy + gx * k;
        for (int dx = 0; dx < k; ++dx) s += xr[dx];
    }
    phi[(size_t)bp * 256 + t] = s * (1.0f / (float)(k * k));

    if (t == 0) {
        float* bb = bbox + (size_t)bp * 4;
        bb[0] = (float)sy;        bb[1] = (float)sx;
        bb[2] = (float)(sy + gh); bb[3] = (float)(sx + gh);
    }
}

// ---------------------------------------------------------------------------
// conv1: implicit GEMM, M = 512*49, N = 32, K = 12*4*4 = 192 (order c,ky,kx).
// One wave per 16x16 tile; output stored [B, 49, 32] (channel-last for conv2).
// ---------------------------------------------------------------------------
__global__ void k_conv1(const float* __restrict__ phi, const float* __restrict__ w,
                        const float* __restrict__ bias, float* __restrict__ h1) {
    const int lane  = threadIdx.x;
    const int tileN = blockIdx.x & 1;
    const int tileM = blockIdx.x >> 1;
    const int lidx  = lane & 15, hi = lane >> 4;

    const int m = tileM * 16 + lidx;        // A-frag row
    const int b = m / 49, s = m % 49, oy = s / 7, ox = s % 7;
    const float* abase = phi + (size_t)b * 3072 + (oy * 2) * 16 + ox * 2;
    const int n = tileN * 16 + lidx;        // B-frag column
    const float* wb = w + (size_t)n * 192;  // conv1_w[n, k] contiguous in k

    v8f acc = {};
    for (int k = 0; k < 192; k += 4) {
        const int k0 = k + hi * 2;          // lanes 0-15: K=0,1 ; 16-31: K=2,3
        const int c = k0 >> 4, r = k0 & 15, ky = r >> 2, kx = r & 3;  // kx even
        v2f a  = *reinterpret_cast<const v2f*>(abase + c * 256 + ky * 16 + kx);
        v2f bv = *reinterpret_cast<const v2f*>(wb + k0);
        acc = wmma_f32(a, bv, acc);
    }
    const float bz = bias[n];
#pragma unroll
    for (int r = 0; r < 8; ++r) {           // D: M = r + 8*hi, N = lidx
        const int mm = tileM * 16 + r + 8 * hi;
        const float v = acc[r] + bz;
        h1[(size_t)mm * 32 + n] = v > 0.0f ? v : 0.0f;
    }
}

// ---------------------------------------------------------------------------
// conv2: implicit GEMM, M = 512*25, N = 64, K = 9*32 = 288 (order kk, c).
// Reads h1 [B,49,32]; writes flattened activations [B, 1600] with the
// reference's c*25 + (y*5+x) ordering.
// ---------------------------------------------------------------------------
__global__ void k_conv2(const float* __restrict__ h1, const float* __restrict__ w,
                        const float* __restrict__ bias, float* __restrict__ h2) {
    const int lane  = threadIdx.x;
    const int tileN = blockIdx.x & 3;
    const int tileM = blockIdx.x >> 2;
    const int lidx  = lane & 15, hi = lane >> 4;

    const int m = tileM * 16 + lidx;
    const int b = m / 25, s = m % 25, oy = s / 5, ox = s % 5;
    const float* abase = h1 + (size_t)b * 1568 + (oy * 7 + ox) * 32;
    const int n = tileN * 16 + lidx;
    const float* wb = w + (size_t)n * 288;  // conv2_w[n, c, ky, kx]

    v8f acc = {};
    for (int k = 0; k < 288; k += 4) {
        const int k0 = k + hi * 2;
        const int kk = k0 >> 5, c = k0 & 31;        // c even -> (c, c+1) valid
        const int ky = kk / 3, kx = kk % 3;
        v2f a  = *reinterpret_cast<const v2f*>(abase + (ky * 7 + kx) * 32 + c);
        v2f bv = { wb[c * 9 + kk], wb[(c + 1) * 9 + kk] };
        acc = wmma_f32(a, bv, acc);
    }
    const float bz = bias[n];
#pragma unroll
    for (int r = 0; r < 8; ++r) {
        const int mm = tileM * 16 + r + 8 * hi;
        const int bb = mm / 25, ss = mm % 25;
        const float v = acc[r] + bz;
        h2[(size_t)bb * 1600 + n * 25 + ss] = v > 0.0f ? v : 0.0f;  // flatten order
    }
}

// ---------------------------------------------------------------------------
// Generic fc + bias + relu: out[M,N] = relu(A[M,K] @ W[N,K]^T + bias)
// ---------------------------------------------------------------------------
__global__ void k_fc_relu(const float* __restrict__ A, const float* __restrict__ W,
                          const float* __restrict__ bias, float* __restrict__ out,
                          int Kdim, int Ndim, int tilesN) {
    const int lane  = threadIdx.x;
    const int tileN = blockIdx.x % tilesN;
    const int tileM = blockIdx.x / tilesN;
    const int lidx  = lane & 15, hi = lane >> 4;

    const float* arow = A + (size_t)(tileM * 16 + lidx) * Kdim;
    const int n = tileN * 16 + lidx;
    const float* wrow = W + (size_t)n * Kdim;

    v8f acc = {};
    for (int k = 0; k < Kdim; k += 4) {
        const int k0 = k + hi * 2;
        v2f a  = *reinterpret_cast<const v2f*>(arow + k0);
        v2f bv = *reinterpret_cast<const v2f*>(wrow + k0);
        acc = wmma_f32(a, bv, acc);
    }
    const float bz = bias[n];
#pragma unroll
    for (int r = 0; r < 8; ++r) {
        const int mm = tileM * 16 + r + 8 * hi;
        const float v = acc[r] + bz;
        out[(size_t)mm * Ndim + n] = v > 0.0f ? v : 0.0f;
    }
}

// ---------------------------------------------------------------------------
// g_t = relu(phi_out @ fc3^T + b3 + l_out @ fc4^T + b4) : both GEMMs fused
// into one accumulator. M = 512, N = 384, K = 256 then 128.
// ---------------------------------------------------------------------------
__global__ void k_fc34(const float* __restrict__ phiOut, const float* __restrict__ W3,
                       const float* __restrict__ b3, const float* __restrict__ lOut,
                       const float* __restrict__ W4, const float* __restrict__ b4,
                       float* __restrict__ gT) {
    const int lane  = threadIdx.x;
    const int tileN = blockIdx.x % 24;
    const int tileM = blockIdx.x / 24;
    const int lidx  = lane & 15, hi = lane >> 4;
    const int m = tileM * 16 + lidx;
    const int n = tileN * 16 + lidx;

    const float* a3 = phiOut + (size_t)m * 256;
    const float* w3 = W3 + (size_t)n * 256;
    const float* a4 = lOut + (size_t)m * 128;
    const float* w4 = W4 + (size_t)n * 128;

    v8f acc = {};
    for (int k = 0; k < 256; k += 4) {
        const int k0 = k + hi * 2;
        v2f a  = *reinterpret_cast<const v2f*>(a3 + k0);
        v2f bv = *reinterpret_cast<const v2f*>(w3 + k0);
        acc = wmma_f32(a, bv, acc);
    }
    for (int k = 0; k < 128; k += 4) {
        const int k0 = k + hi * 2;
        v2f a  = *reinterpret_cast<const v2f*>(a4 + k0);
        v2f bv = *reinterpret_cast<const v2f*>(w4 + k0);
        acc = wmma_f32(a, bv, acc);
    }
    const float bz = b3[n] + b4[n];
#pragma unroll
    for (int r = 0; r < 8; ++r) {
        const int mm = tileM * 16 + r + 8 * hi;
        const float v = acc[r] + bz;
        gT[(size_t)mm * 384 + n] = v > 0.0f ? v : 0.0f;
    }
}

// ---------------------------------------------------------------------------
extern "C" void kernel_launch(void* const* d_in, const int* in_sizes, int n_in,
                              void* d_out, int out_size, void* d_ws, size_t ws_size,
                              hipStream_t stream) {
    (void)in_sizes; (void)n_in; (void)out_size; (void)ws_size;

    const float* x   = (const float*)d_in[0];   // [512,1,256,256]
    const float* l   = (const float*)d_in[1];   // [512,4,2]
    const float* c1w = (const float*)d_in[2];   // [32,12,4,4]
    const float* c1b = (const float*)d_in[3];   // [32]
    const float* c2w = (const float*)d_in[4];   // [64,32,3,3]
    const float* c2b = (const float*)d_in[5];   // [64]
    const float* f1w = (const float*)d_in[6];   // [256,1600]
    const float* f1b = (const float*)d_in[7];   // [256]
    const float* f2w = (const float*)d_in[8];   // [128,8]
    const float* f2b = (const float*)d_in[9];   // [128]
    const float* f3w = (const float*)d_in[10];  // [384,256]
    const float* f3b = (const float*)d_in[11];  // [384]
    const float* f4w = (const float*)d_in[12];  // [384,128]
    const float* f4b = (const float*)d_in[13];  // [384]

    float* out    = (float*)d_out;
    float* phiOut = out;                        // [512,256]
    float* gT     = out + 512 * 256;            // [512,384]
    float* bbox   = out + 512 * 256 + 512 * 384;// [512,12,4]

    float* ws   = (float*)d_ws;
    float* phi  = ws;                           // 512*12*256
    float* h1   = phi + 512 * 12 * 256;         // 512*49*32  ([B,49,32])
    float* h2   = h1 + 512 * 49 * 32;           // 512*1600
    float* lout = h2 + 512 * 1600;              // 512*128

    k_foveate<<<512 * 12, 256, 0, stream>>>(x, l, phi, bbox);
    k_conv1 <<<1568 * 2, 32, 0, stream>>>(phi, c1w, c1b, h1);
    k_conv2 <<<800 * 4, 32, 0, stream>>>(h1, c2w, c2b, h2);
    k_fc_relu<<<32 * 16, 32, 0, stream>>>(h2, f1w, f1b, phiOut, 1600, 256, 16);
    k_fc_relu<<<32 * 8,  32, 0, stream>>>(l,  f2w, f2b, lout,   8,    128, 8);
    k_fc34  <<<32 * 24, 32, 0, stream>>>(phiOut, f3w, f3b, lout, f4w, f4b, gT);
}